// tensorFusion_60430189855099
// MI455X (gfx1250) — compile-verified
//
#include <hip/hip_runtime.h>

// CDNA5 / gfx1250, wave32.
typedef __attribute__((ext_vector_type(2))) float v2f;
typedef __attribute__((ext_vector_type(8))) float v8f;

// out[b, i, j, k] = ah[b,i] * vh[b,j] * th[b,k],  i,j,k in [0,64)
// ah = [1, a[b,:]], vh = [1, v[b,:]], th = [1, t[b,:]]
// Block = (batch b, i-tile ti of 16 rows). Output region per block:
// 16*64*64 floats = 256 KB, contiguous.
__global__ __launch_bounds__(256) void tensorFusion_60430189855099_kernel(
    const float* __restrict__ t,   // [B, 63]
    const float* __restrict__ a,   // [B, 63]
    const float* __restrict__ v,   // [B, 63]
    float* __restrict__ out)       // [B, 64*64*64]
{
    constexpr int D = 63;
    const int b   = blockIdx.x >> 2;   // batch index
    const int ti  = blockIdx.x & 3;    // which 16-row i-tile
    const int tid = threadIdx.x;

    __shared__ alignas(16) float ah_s[16];       // padded a, rows ti*16 .. ti*16+15
    __shared__ alignas(16) float vh_s[64];       // padded v
    __shared__ alignas(16) float th_s[64];       // padded t
    __shared__ alignas(16) float M_s[16 * 64];   // M[m][n] = ah[m] * vh[n]

    // ---- stage padded input vectors into LDS ----
    if (tid < 64) {
        vh_s[tid] = (tid == 0) ? 1.0f : v[b * D + tid - 1];
        th_s[tid] = (tid == 0) ? 1.0f : t[b * D + tid - 1];
    } else if (tid < 80) {
        const int m = tid - 64;
        const int i = ti * 16 + m;
        ah_s[m] = (i == 0) ? 1.0f : a[b * D + i - 1];
    }
    __syncthreads();

    // ---- rank-1 outer product via V_WMMA_F32_16X16X4_F32 ----
    // Waves 0..3: wave w computes the 16x16 tile M[m][w*16 + n].
    // A (16x4 f32): lanes 0-15 hold K=0 (VGPR0) / K=1 (VGPR1);
    //               lanes 16-31 hold K=2 / K=3. Only K=0 is nonzero (= ah).
    // B (4x16 f32): row K=0 striped over lanes 0-15 in VGPR0 (= vh); rest 0.
    // D (16x16 f32, 8 VGPRs): VGPR r -> M = r + 8*(lane>=16), N = lane&15.
    const int wave = tid >> 5;
    const int lane = tid & 31;
    if (wave < 4) {                    // uniform per wave: EXEC all-ones inside
        const int half = lane >> 4;    // 0: lanes 0-15, 1: lanes 16-31
        const int l16  = lane & 15;

        v2f A;
        A.x = half ? 0.0f : ah_s[l16];            // K=0 column = ah
        A.y = 0.0f;
        v2f Bm;
        Bm.x = half ? 0.0f : vh_s[wave * 16 + l16]; // K=0 row = vh
        Bm.y = 0.0f;

        v8f C = {};
        // (neg_a, A, neg_b, B, c_mod, C, reuse_a, reuse_b)
        C = __builtin_amdgcn_wmma_f32_16x16x4_f32(
                false, A, false, Bm, (short)0, C, false, false);

        #pragma unroll
        for (int r = 0; r < 8; ++r) {
            const int m = r + 8 * half;
            M_s[m * 64 + wave * 16 + l16] = C[r];
        }
    }
    __syncthreads();

    // ---- bandwidth phase: out[row, k] = M_s[row] * th[k] ----
    // row = m*64 + j in [0,1024); k handled as float4 quads.
    // Thread: fixed k-quad (tid&15), row = (tid>>4) + 16*it.
    // A wave stores 512 contiguous bytes per iteration (fully coalesced b128).
    const int    k4    = tid & 15;
    const int    rbase = tid >> 4;
    const float4 th4   = ((const float4*)th_s)[k4];  // constant per thread

    float4* __restrict__ out4 =
        (float4*)(out + (size_t)b * (64 * 64 * 64) + (size_t)ti * (16 * 64 * 64));

    #pragma unroll 4
    for (int it = 0; it < 64; ++it) {
        const int   row = rbase + it * 16;   // 0..1023
        const float mv  = M_s[row];          // LDS broadcast within half-wave
        float4 val;
        val.x = mv * th4.x;
        val.y = mv * th4.y;
        val.z = mv * th4.z;
        val.w = mv * th4.w;
        out4[row * 16 + k4] = val;           // global_store_b128
    }
}

extern "C" void kernel_launch(void* const* d_in, const int* in_sizes, int n_in,
                              void* d_out, int out_size, void* d_ws, size_t ws_size,
                              hipStream_t stream) {
    // setup_inputs() order: "t", "a", "v" — all float32 [B, 63]
    const float* t = (const float*)d_in[0];
    const float* a = (const float*)d_in[1];
    const float* v = (const float*)d_in[2];
    float* out = (float*)d_out;

    const int B = in_sizes[0] / 63;          // 128
    dim3 grid(B * 4), block(256);
    hipLaunchKernelGGL(tensorFusion_60430189855099_kernel,
                       grid, block, 0, stream, t, a, v, out);
}